// Navier_Stokes_informed_loss_19576460935734
// MI455X (gfx1250) — compile-verified
//
#include <hip/hip_runtime.h>
#include <hip/hip_bf16.h>
#include <stdint.h>

// ---------------------------------------------------------------------------
// Navier-Stokes physics-informed loss, MI455X (gfx1250).
// Memory-bound stencil: TDM (tensor_load_to_lds) stages halo tiles into LDS,
// VALU does fp32 FD math, deterministic wave32 shuffle + 2-pass reduction.
// ---------------------------------------------------------------------------

typedef __attribute__((ext_vector_type(4))) unsigned int u32x4;
typedef __attribute__((ext_vector_type(8))) int          i32x8;
typedef __attribute__((ext_vector_type(4))) int          i32x4;

#define TH 16
#define TW 64
#define HP (TH + 2)          // 18
#define WP (TW + 2)          // 66
#define WDIM 512
#define HDIM 512
#define PLANE (512 * 512)
#define NB 62                // interior batches
#define NHT 32               // ceil(510/16)
#define NWT 8                // ceil(510/64)
#define NPART (NB * NHT * NWT)
#define NPTS 16126200.0f     // 62*510*510

// Issue one 2-D TDM tile load: global (remW x remH window, row stride 512
// elems) -> contiguous LDS tile of tileW x tileH fp32. Out-of-window reads
// are zero-filled by the TDM (tensor_dim = remaining extent from tile start).
__device__ __forceinline__ void tdm_load_tile(const void* gsrc, const void* ldst,
                                              unsigned tileW, unsigned tileH,
                                              unsigned remW, unsigned remH)
{
  uint64_t ga = (uint64_t)(uintptr_t)gsrc;
  uint32_t la = (uint32_t)(uintptr_t)ldst;   // low 32 bits of flat addr = LDS offset

  u32x4 g0;
  g0[0] = 1u;                                          // count=1, user descriptor
  g0[1] = la;                                          // lds_addr (bytes)
  g0[2] = (uint32_t)ga;                                // global_addr[31:0]
  g0[3] = (uint32_t)((ga >> 32) & 0x01FFFFFFu)         // global_addr[56:32]
        | (2u << 30);                                  // type = 2 ("image")

  i32x8 g1;
  g1[0] = 0x00020000;                                  // data_size = 2 (4 bytes)
  g1[1] = (int)((remW & 0xFFFFu) << 16);               // tensor_dim0[15:0]
  g1[2] = (int)(((remW >> 16) & 0xFFFFu)               // tensor_dim0[31:16]
        | ((remH & 0xFFFFu) << 16));                   // tensor_dim1[15:0]
  g1[3] = (int)(((remH >> 16) & 0xFFFFu)               // tensor_dim1[31:16]
        | ((tileW & 0xFFFFu) << 16));                  // tile_dim0
  g1[4] = (int)(tileH & 0xFFFFu);                      // tile_dim1 (tile_dim2=0)
  g1[5] = WDIM;                                        // tensor_dim0_stride[31:0]
  g1[6] = 0;
  g1[7] = 0;

  i32x4 z4 = {0, 0, 0, 0};                             // groups 2/3 unused (2-D)
  i32x8 z8 = {0, 0, 0, 0, 0, 0, 0, 0};
  __builtin_amdgcn_tensor_load_to_lds(g0, g1, z4, z4, z8, 0);
}

// FD residuals at one interior point, from LDS tiles (values scaled by std).
__device__ __forceinline__ void residuals(const float* __restrict__ U,
                                          const float* __restrict__ V,
                                          const float* __restrict__ P,
                                          const float* __restrict__ Um,
                                          const float* __restrict__ Up,
                                          const float* __restrict__ Vm,
                                          const float* __restrict__ Vp,
                                          int hl, int wl, float sd,
                                          float& r0, float& r1, float& r2)
{
  const int c  = (hl + 1) * WP + (wl + 1);   // halo-tile center
  const int cc = hl * TW + wl;               // center-tile index

  const float u   = U[c]      * sd, v   = V[c]      * sd;
  const float uxp = U[c + WP] * sd, uxm = U[c - WP] * sd;   // h+1 / h-1
  const float uyp = U[c + 1]  * sd, uym = U[c - 1]  * sd;   // w+1 / w-1
  const float vxp = V[c + WP] * sd, vxm = V[c - WP] * sd;
  const float vyp = V[c + 1]  * sd, vym = V[c - 1]  * sd;
  const float pxp = P[c + WP] * sd, pxm = P[c - WP] * sd;
  const float pyp = P[c + 1]  * sd, pym = P[c - 1]  * sd;
  const float utp = Up[cc]    * sd, utm = Um[cc]    * sd;
  const float vtp = Vp[cc]    * sd, vtm = Vm[cc]    * sd;

  const float i2d  = 127.5f;    // 1/(2*DX) = 1/(2*DY)
  const float idd  = 65025.0f;  // 1/DX^2   = 1/DY^2
  const float idt  = 32.0f;     // 1/DT
  const float MU_  = 1e-5f;

  const float dudx = (uxp - uxm) * i2d, dudy = (uyp - uym) * i2d;
  const float dvdx = (vxp - vxm) * i2d, dvdy = (vyp - vym) * i2d;
  const float dpdx = (pxp - pxm) * i2d, dpdy = (pyp - pym) * i2d;
  const float dudt = (utp - utm) * idt, dvdt = (vtp - vtm) * idt;
  const float lapu = (uxp - 2.0f * u + uxm) * idd + (uyp - 2.0f * u + uym) * idd;
  const float lapv = (vxp - 2.0f * v + vxm) * idd + (vyp - 2.0f * v + vym) * idd;

  r0 = fabsf(dudx + dvdy);
  r1 = fabsf((dudt + u * dudx + v * dudy) + dpdx - MU_ * lapu);  // RHO = 1
  r2 = fabsf((dvdt + u * dvdx + v * dvdy) + dpdy - MU_ * lapv);
}

__global__ __launch_bounds__(256, 1)
void ns_stage1(const float* __restrict__ xin, const float* __restrict__ yin,
               const float* __restrict__ stdp, float* __restrict__ partial)
{
  __shared__ float sHalo[2][3][HP * WP];   // [x|y][u,v,p] halo tiles at batch b
  __shared__ float sTim[2][4][TH * TW];    // [x|y][u@b-1,u@b+1,v@b-1,v@b+1]
  __shared__ float sRed[24];               // 8 waves x 3 partials

  const int tid = (int)threadIdx.x;
  const int b   = 1 + (int)blockIdx.z;
  const int h0  = 1 + (int)blockIdx.y * TH;
  const int w0  = 1 + (int)blockIdx.x * TW;

  // Wave 0 (uniform branch) issues all 14 TDM tile loads and waits on them.
  if (tid < 32) {
    const unsigned remWH = (unsigned)(WDIM - (w0 - 1));
    const unsigned remHH = (unsigned)(HDIM - (h0 - 1));
    const unsigned remWC = (unsigned)(WDIM - w0);
    const unsigned remHC = (unsigned)(HDIM - h0);
#pragma unroll
    for (int a = 0; a < 2; ++a) {
      const float* base = a ? yin : xin;
      const size_t boff = (size_t)b * 3 * PLANE;
#pragma unroll
      for (int ch = 0; ch < 3; ++ch) {
        const float* src = base + boff + (size_t)ch * PLANE
                         + (size_t)(h0 - 1) * WDIM + (size_t)(w0 - 1);
        tdm_load_tile(src, sHalo[a][ch], WP, HP, remWH, remHH);
      }
      const size_t coff = (size_t)h0 * WDIM + (size_t)w0;
      const float* um = base + (size_t)(b - 1) * 3 * PLANE + 0 * PLANE + coff;
      const float* up = base + (size_t)(b + 1) * 3 * PLANE + 0 * PLANE + coff;
      const float* vm = base + (size_t)(b - 1) * 3 * PLANE + 1 * PLANE + coff;
      const float* vp = base + (size_t)(b + 1) * 3 * PLANE + 1 * PLANE + coff;
      tdm_load_tile(um, sTim[a][0], TW, TH, remWC, remHC);
      tdm_load_tile(up, sTim[a][1], TW, TH, remWC, remHC);
      tdm_load_tile(vm, sTim[a][2], TW, TH, remWC, remHC);
      tdm_load_tile(vp, sTim[a][3], TW, TH, remWC, remHC);
    }
    __builtin_amdgcn_s_wait_tensorcnt(0);
  }
  __syncthreads();

  const float sd = stdp[0];
  float s0 = 0.0f, s1 = 0.0f, s2 = 0.0f;

  const int wl = tid & (TW - 1);
  const int hb = tid >> 6;                 // 0..3
#pragma unroll
  for (int k = 0; k < 4; ++k) {
    const int hl = hb + 4 * k;             // 0..15
    const int h = h0 + hl, w = w0 + wl;
    if (h <= 510 && w <= 510) {
      float xr0, xr1, xr2, yr0, yr1, yr2;
      residuals(sHalo[0][0], sHalo[0][1], sHalo[0][2],
                sTim[0][0], sTim[0][1], sTim[0][2], sTim[0][3],
                hl, wl, sd, xr0, xr1, xr2);
      residuals(sHalo[1][0], sHalo[1][1], sHalo[1][2],
                sTim[1][0], sTim[1][1], sTim[1][2], sTim[1][3],
                hl, wl, sd, yr0, yr1, yr2);
      const float d0 = yr0 - xr0, d1 = yr1 - xr1, d2 = yr2 - xr2;
      s0 += d0 * d0; s1 += d1 * d1; s2 += d2 * d2;
    }
  }

  // Deterministic wave32 tree reduction.
#pragma unroll
  for (int off = 16; off > 0; off >>= 1) {
    s0 += __shfl_down(s0, (unsigned)off, 32);
    s1 += __shfl_down(s1, (unsigned)off, 32);
    s2 += __shfl_down(s2, (unsigned)off, 32);
  }
  const int wave = tid >> 5, lane = tid & 31;
  if (lane == 0) {
    sRed[wave * 3 + 0] = s0;
    sRed[wave * 3 + 1] = s1;
    sRed[wave * 3 + 2] = s2;
  }
  __syncthreads();
  if (tid == 0) {
    float t0 = 0.0f, t1 = 0.0f, t2 = 0.0f;
#pragma unroll
    for (int i = 0; i < 8; ++i) {
      t0 += sRed[i * 3 + 0]; t1 += sRed[i * 3 + 1]; t2 += sRed[i * 3 + 2];
    }
    const size_t bi = ((size_t)blockIdx.z * gridDim.y + blockIdx.y) * gridDim.x
                    + blockIdx.x;
    partial[bi * 3 + 0] = t0;
    partial[bi * 3 + 1] = t1;
    partial[bi * 3 + 2] = t2;
  }
}

__global__ __launch_bounds__(256, 1)
void ns_stage2(const float* __restrict__ partial, int n, float* __restrict__ out)
{
  __shared__ float r0[256], r1[256], r2[256];
  const int tid = (int)threadIdx.x;
  float s0 = 0.0f, s1 = 0.0f, s2 = 0.0f;
  for (int i = tid; i < n; i += 256) {                 // fixed order -> deterministic
    s0 += partial[(size_t)i * 3 + 0];
    s1 += partial[(size_t)i * 3 + 1];
    s2 += partial[(size_t)i * 3 + 2];
  }
  r0[tid] = s0; r1[tid] = s1; r2[tid] = s2;
  __syncthreads();
  for (int s = 128; s > 0; s >>= 1) {
    if (tid < s) { r0[tid] += r0[tid + s]; r1[tid] += r1[tid + s]; r2[tid] += r2[tid + s]; }
    __syncthreads();
  }
  if (tid == 0) {
    const float invN = 1.0f / NPTS;
    out[0] = 0.001f * (r0[0] + r1[0] + r2[0]) * invN;
  }
}

extern "C" void kernel_launch(void* const* d_in, const int* in_sizes, int n_in,
                              void* d_out, int out_size, void* d_ws, size_t ws_size,
                              hipStream_t stream) {
  const float* x    = (const float*)d_in[0];
  const float* y    = (const float*)d_in[1];
  const float* stdp = (const float*)d_in[2];
  float* out        = (float*)d_out;
  float* partial    = (float*)d_ws;   // NPART*3 floats = 190 KB

  dim3 grid(NWT, NHT, NB);            // 8 x 32 x 62 = 15872 blocks
  ns_stage1<<<grid, 256, 0, stream>>>(x, y, stdp, partial);
  ns_stage2<<<1, 256, 0, stream>>>(partial, NPART, out);
}